// CochainMessagePassing_79688823210736
// MI455X (gfx1250) — compile-verified
//
#include <hip/hip_runtime.h>

// CochainMessagePassing attention-scores kernel for gfx1250 (MI455X).
//
// Per (b, n, h):  e_pair = a2 @ xh^T  (M=1024, N=1024, K=64, fp32)
//                 scores = e_self[:,None] + e_pair ; softmax over j
// Output layout:  out[b, n, i, h, j]  (512 MB fp32 -> write-bandwidth bound)
//
// GEMM on V_WMMA_F32_16X16X4_F32 (exact fp32 matrix pipe).  xh tiles are
// streamed into LDS with GLOBAL_LOAD_ASYNC_TO_LDS_B128 (ASYNCcnt-tracked,
// in-order completion) and double-buffered so the next chunk's HBM latency
// hides under the current chunk's 16 chained WMMAs.  Output uses
// non-temporal stores so the 512 MB write stream does not evict the
// heavily-reused 96 MB of inputs from the 192 MB L2.

#define NB_N 4
#define NCC  1024
#define DD   512
#define HH   8
#define DH   64

typedef float v2f __attribute__((ext_vector_type(2)));
typedef float v8f __attribute__((ext_vector_type(8)));

#define XS_STRIDE 68   // 68 % 64 == 4 -> conflict-free b64 LDS fragment reads
#define A2_STRIDE 68

// Byte offset of a static __shared__ object within the workgroup LDS
// allocation (addrspace(3) pointer value == LDS byte offset).
__device__ __forceinline__ uint32_t lds_addr_of(void* p) {
    return (uint32_t)(uintptr_t)(__attribute__((address_space(3))) void*)p;
}

// Async global -> LDS copy, 16B per lane.  VDST operand = per-lane LDS byte
// offset, VADDR = per-lane 64-bit global address (GV mode).  ASYNCcnt-tracked.
__device__ __forceinline__ void async_copy_b128(uint32_t lds_off, const float* gp) {
    asm volatile("global_load_async_to_lds_b128 %0, %1, off"
                 :: "v"(lds_off), "v"(gp) : "memory");
}

#define WAIT_ASYNCCNT(n) asm volatile("s_wait_asynccnt " #n ::: "memory")

__global__ __launch_bounds__(256)
void cochain_attn_kernel(const float* __restrict__ x,
                         const float* __restrict__ attn_w,
                         float* __restrict__ out)
{
    __shared__ float A2s[16 * A2_STRIDE];       //  4352 B : a2 tile (16 x 64)
    __shared__ float Xs[2][128 * XS_STRIDE];    // 2x34816 B : xh chunk, dbl-buf
    __shared__ float Es[16];                    // e_self per i-row
    __shared__ float Ps[256];                   // e_self partials
    __shared__ float WR[8 * 16];                // cross-wave softmax reductions

    const int t    = threadIdx.x;
    const int lane = t & 31;
    const int wave = t >> 5;

    int wg = blockIdx.x;
    const int it = wg & 63;  wg >>= 6;   // i-tile (16 rows)
    const int h  = wg & 7;   wg >>= 3;
    const int n  = wg & 3;   wg >>= 2;
    const int b  = wg;

    const int  i_base = it * 16;
    const long bn     = (long)(b * NB_N + n);

    const float* xh_base = x + bn * NCC * DD + h * DH;               // + j*DD
    const float* aw_base = attn_w + (bn * HH + h) * (long)NCC * 128; // + i*128

    // Per-thread (row, k4) assignment used by all LDS staging
    const int srow = t >> 4;          // 0..15
    const int sk4  = (t & 15) * 4;    // 0..60

    // ---- Phase 0: stage a2 tile into LDS; compute e_self[16] ----
    {
        const float* awr = aw_base + (long)(i_base + srow) * 128;
        float4 a2 = *(const float4*)(awr + 64 + sk4);
        float4 a1 = *(const float4*)(awr + sk4);
        float4 xv = *(const float4*)(xh_base + (long)(i_base + srow) * DD + sk4);
        float* d = &A2s[srow * A2_STRIDE + sk4];
        *(float2*)(d)     = make_float2(a2.x, a2.y);
        *(float2*)(d + 2) = make_float2(a2.z, a2.w);
        Ps[t] = a1.x * xv.x + a1.y * xv.y + a1.z * xv.z + a1.w * xv.w;
    }
    __syncthreads();
    if (t < 16) {
        float s = 0.f;
        #pragma unroll
        for (int k = 0; k < 16; ++k) s += Ps[t * 16 + k];
        Es[t] = s;
    }
    __syncthreads();

    // ---- Preload A fragments (constant across the whole j loop) ----
    // ISA A-layout (32-bit 16x4): lanes 0-15 hold M=lane, K=kb+0 / kb+1 in
    // v0/v1; lanes 16-31 hold M=lane-16, K=kb+2 / kb+3.
    const int mrow  = lane & 15;
    const int khalf = (lane < 16) ? 0 : 2;
    float afrag0[16], afrag1[16];
    #pragma unroll
    for (int ks = 0; ks < 16; ++ks) {
        const int kk = ks * 4 + khalf;
        float2 a = *(const float2*)&A2s[mrow * A2_STRIDE + kk];
        afrag0[ks] = a.x;
        afrag1[ks] = a.y;
    }

    // C/D layout: element r of v8f at lane l holds M = r + 8*(l>=16), N = l%16.
    // Seed the accumulator with e_self[M] (constant along N => softmax-safe).
    const int half8 = (lane < 16) ? 0 : 8;
    v8f esv;
    #pragma unroll
    for (int r = 0; r < 8; ++r) esv[r] = Es[half8 + r];

    v8f accs[8];  // one 16x16 score tile per j-chunk, kept in VGPRs

    // Async-copy issue for one 128-row xh chunk (8 x b128 per thread)
    const uint32_t xs_off[2] = { lds_addr_of(&Xs[0][0]), lds_addr_of(&Xs[1][0]) };
    auto issue_chunk = [&](int jc, int buf) {
        #pragma unroll
        for (int itr = 0; itr < 8; ++itr) {
            const int u   = t + itr * 256;
            const int row = u >> 4;      // 0..127
            const int k4  = (u & 15) * 4;
            async_copy_b128(xs_off[buf] + (uint32_t)(row * XS_STRIDE + k4) * 4u,
                            xh_base + (long)(jc * 128 + row) * DD + k4);
        }
    };

    // ---- Main loop: j in chunks of 128 (8 waves x 16 cols), double-buffered ----
    issue_chunk(0, 0);
    for (int jc = 0; jc < 8; ++jc) {
        if (jc < 7) {
            issue_chunk(jc + 1, (jc + 1) & 1);
            WAIT_ASYNCCNT(8);   // async loads complete in order: drains chunk jc
        } else {
            WAIT_ASYNCCNT(0);
        }
        __syncthreads();

        // B (4x16, K x N) fragment: B[k][j] = xh[j_base + j][k].
        // f32 B layout is not tabulated in the ISA; we mirror A's K split:
        // lanes 0-15: N=lane, K=kb+0 / kb+1; lanes 16-31: N=lane-16, K=kb+2/3.
        const float* xsb = &Xs[jc & 1][0];
        v8f acc = esv;
        const int jrow = wave * 16 + (lane & 15);
        #pragma unroll
        for (int ks = 0; ks < 16; ++ks) {
            const int kk = ks * 4 + khalf;
            float2 bv = *(const float2*)&xsb[jrow * XS_STRIDE + kk];
            v2f a;  a[0] = afrag0[ks]; a[1] = afrag1[ks];
            v2f bb; bb[0] = bv.x;      bb[1] = bv.y;
            acc = __builtin_amdgcn_wmma_f32_16x16x4_f32(
                false, a, false, bb, (short)0, acc, false, false);
        }
        accs[jc] = acc;
        __syncthreads();   // all waves done reading before buffer is re-filled
    }

    // ---- Softmax over j (1024) : registers + shfl_xor + tiny LDS ----
    float gmax[8], gsum[8];

    // row max: local over 8 chunks, then across the 16 lanes of this half
    #pragma unroll
    for (int r = 0; r < 8; ++r) {
        float m = accs[0][r];
        #pragma unroll
        for (int jc = 1; jc < 8; ++jc) m = fmaxf(m, accs[jc][r]);
        #pragma unroll
        for (int off = 1; off < 16; off <<= 1)
            m = fmaxf(m, __shfl_xor(m, off, 32));
        gmax[r] = m;
    }
    if (lane == 0 || lane == 16) {
        #pragma unroll
        for (int r = 0; r < 8; ++r) WR[wave * 16 + half8 + r] = gmax[r];
    }
    __syncthreads();
    #pragma unroll
    for (int r = 0; r < 8; ++r) {
        float m = WR[half8 + r];
        #pragma unroll
        for (int w2 = 1; w2 < 8; ++w2) m = fmaxf(m, WR[w2 * 16 + half8 + r]);
        gmax[r] = m;
    }
    __syncthreads();   // WR reused for sums

    // exp in place + row sum
    #pragma unroll
    for (int r = 0; r < 8; ++r) {
        float s = 0.f;
        #pragma unroll
        for (int jc = 0; jc < 8; ++jc) {
            float e = __expf(accs[jc][r] - gmax[r]);
            accs[jc][r] = e;
            s += e;
        }
        #pragma unroll
        for (int off = 1; off < 16; off <<= 1)
            s += __shfl_xor(s, off, 32);
        gsum[r] = s;
    }
    if (lane == 0 || lane == 16) {
        #pragma unroll
        for (int r = 0; r < 8; ++r) WR[wave * 16 + half8 + r] = gsum[r];
    }
    __syncthreads();
    #pragma unroll
    for (int r = 0; r < 8; ++r) {
        float s = 0.f;
        #pragma unroll
        for (int w2 = 0; w2 < 8; ++w2) s += WR[w2 * 16 + half8 + r];
        gsum[r] = 1.0f / s;
    }

    // ---- Write out[b, n, i_base+i, h, j] (contiguous in j) ----
    // Non-temporal: write-once 512 MB stream must not evict L2-resident inputs.
    const int jl = lane & 15;
    #pragma unroll
    for (int r = 0; r < 8; ++r) {
        const int i = half8 + r;
        float* orow = out + (((bn * NCC + i_base + i) * HH + h) * (long)NCC);
        #pragma unroll
        for (int jc = 0; jc < 8; ++jc)
            __builtin_nontemporal_store(accs[jc][r] * gsum[r],
                                        &orow[jc * 128 + wave * 16 + jl]);
    }
}

extern "C" void kernel_launch(void* const* d_in, const int* in_sizes, int n_in,
                              void* d_out, int out_size, void* d_ws, size_t ws_size,
                              hipStream_t stream) {
    const float* x      = (const float*)d_in[0];
    const float* attn_w = (const float*)d_in[1];
    float* out = (float*)d_out;
    (void)in_sizes; (void)n_in; (void)out_size; (void)d_ws; (void)ws_size;

    // blocks: B * N_NB * H * (N_C/16) = 4*4*8*64 = 8192
    dim3 grid(8192), block(256);
    cochain_attn_kernel<<<grid, block, 0, stream>>>(x, attn_w, out);
}